// Net_33432025432567
// MI455X (gfx1250) — compile-verified
//
#include <hip/hip_runtime.h>

typedef __attribute__((ext_vector_type(2))) float v2f;
typedef __attribute__((ext_vector_type(8))) float v8f;

#define THREADS 256

// Native CDNA5 no-return fp32 atomic add (STOREcnt-tracked), device scope so the
// RMW lands at L2 (coherence point for cross-WGP accumulation). s_endpgm's
// implicit wait-idle drains outstanding STOREcnt at kernel end.
__device__ __forceinline__ void atomic_add_f32(float* p, float v) {
    asm volatile("global_atomic_add_f32 %0, %1, off scope:SCOPE_DEV"
                 :
                 : "v"(p), "v"(v)
                 : "memory");
}

// ---- degree / normalization prep -------------------------------------------

__global__ void k_init_deg(float* __restrict__ deg, int n) {
    int i = blockIdx.x * blockDim.x + threadIdx.x;
    if (i < n) deg[i] = 1.0f;                       // self-loop
}

__global__ void k_count_deg(const int* __restrict__ dst, float* __restrict__ deg, int e) {
    int i = blockIdx.x * blockDim.x + threadIdx.x;
    if (i < e) atomic_add_f32(&deg[dst[i]], 1.0f);
}

__global__ void k_inv(float* __restrict__ deg_inv, int n) {
    int i = blockIdx.x * blockDim.x + threadIdx.x;
    if (i < n) deg_inv[i] = rsqrtf(deg_inv[i]);     // in place: deg -> deg^{-1/2}
}

__global__ void k_edge_norm(const int* __restrict__ src, const int* __restrict__ dst,
                            const float* __restrict__ inv, float* __restrict__ norm, int e) {
    int i = blockIdx.x * blockDim.x + threadIdx.x;
    if (i < e) norm[i] = inv[src[i]] * inv[dst[i]];
}

// ---- WMMA fp32 GEMM: C[M x NOUT] = A[M x 64] * W[64 x NOUT] (+bias) --------
// One wave per 16-row tile. K=64 -> 16 chained V_WMMA_F32_16X16X4_F32.
// Fragment layouts per CDNA5 ISA 7.12.2:
//   A 16x4 f32 : lanes 0-15 hold (M=lane, K=0..1) in v0..v1; lanes 16-31 hold K=2..3
//   B 4x16 f32 : lanes 0-15 hold (N=lane, K=0..1); lanes 16-31 hold K=2..3
//   C/D 16x16  : vgpr j, lanes 0-15 -> (M=j, N=lane); lanes 16-31 -> (M=j+8, N=lane-16)
template <int NOUT, bool BIAS>
__global__ void k_gemm_wmma(const float* __restrict__ A, const float* __restrict__ W,
                            const float* __restrict__ bias, float* __restrict__ C, int M) {
    int wave = (blockIdx.x * blockDim.x + threadIdx.x) >> 5;
    int lane = threadIdx.x & 31;
    int m0 = wave * 16;
    if (m0 >= M) return;                            // wave-uniform: EXEC stays all-1s

    const int half  = lane >> 4;                    // 0 or 1
    const int idx16 = lane & 15;
    const int khi   = half << 1;                    // 0 or 2

    // Load the 16x64 A tile once; reuse across all N-tiles.
    v2f afrag[16];
    const float* arow = A + (size_t)(m0 + idx16) * 64;
#pragma unroll
    for (int kk = 0; kk < 16; ++kk) {
        afrag[kk].x = arow[kk * 4 + khi];
        afrag[kk].y = arow[kk * 4 + khi + 1];
    }

#pragma unroll
    for (int nt = 0; nt < NOUT / 16; ++nt) {
        const int n0 = nt * 16;
        v8f acc = {};
#pragma unroll
        for (int kk = 0; kk < 16; ++kk) {
            v2f b;
            b.x = W[(size_t)(kk * 4 + khi) * NOUT + n0 + idx16];
            b.y = W[(size_t)(kk * 4 + khi + 1) * NOUT + n0 + idx16];
            acc = __builtin_amdgcn_wmma_f32_16x16x4_f32(
                false, afrag[kk], false, b, (short)0, acc, false, false);
        }
        const float badd = BIAS ? bias[n0 + idx16] : 0.0f;
#pragma unroll
        for (int j = 0; j < 8; ++j) {
            int row = m0 + j + (half << 3);
            C[(size_t)row * NOUT + n0 + idx16] = acc[j] + badd;
        }
    }
}

// ---- self-loop init: agg = hlin * deg^{-1} ---------------------------------
__global__ void k_self_init(const float* __restrict__ hlin, const float* __restrict__ inv,
                            float* __restrict__ agg, int n16) {
    int i = blockIdx.x * blockDim.x + threadIdx.x;
    if (i >= n16) return;
    int node = i >> 4;
    float w = inv[node];
    w *= w;                                         // deg^{-1}
    float4 v = ((const float4*)hlin)[i];
    v.x *= w; v.y *= w; v.z *= w; v.w *= w;
    ((float4*)agg)[i] = v;
}

// ---- edge scatter: agg[dst] += hlin[src] * norm_e --------------------------
// 16 lanes per edge, float4 per lane: coalesced 256B gather per edge; the
// whole h matrix (25.6 MB) is L2-resident on MI455X's 192 MB L2.
__global__ void k_scatter(const int* __restrict__ src, const int* __restrict__ dst,
                          const float* __restrict__ norm, const float* __restrict__ hlin,
                          float* __restrict__ agg, int e) {
    int t  = blockIdx.x * blockDim.x + threadIdx.x;
    int ei = t >> 4;
    int c  = t & 15;
    if (ei >= e) return;
    int   s = src[ei];
    int   d = dst[ei];
    float w = norm[ei];
    float4 v = ((const float4*)hlin)[(size_t)s * 16 + c];
    float* p = agg + (size_t)d * 64 + (size_t)c * 4;
    atomic_add_f32(p + 0, v.x * w);
    atomic_add_f32(p + 1, v.y * w);
    atomic_add_f32(p + 2, v.z * w);
    atomic_add_f32(p + 3, v.w * w);
}

// ---- bias + ReLU (in place) ------------------------------------------------
__global__ void k_bias_relu(float* __restrict__ h, const float* __restrict__ b, int n16) {
    int i = blockIdx.x * blockDim.x + threadIdx.x;
    if (i >= n16) return;
    int f = (i & 15) << 2;
    float4 v = ((float4*)h)[i];
    v.x = fmaxf(v.x + b[f + 0], 0.0f);
    v.y = fmaxf(v.y + b[f + 1], 0.0f);
    v.z = fmaxf(v.z + b[f + 2], 0.0f);
    v.w = fmaxf(v.w + b[f + 3], 0.0f);
    ((float4*)h)[i] = v;
}

// ---------------------------------------------------------------------------

extern "C" void kernel_launch(void* const* d_in, const int* in_sizes, int n_in,
                              void* d_out, int out_size, void* d_ws, size_t ws_size,
                              hipStream_t stream) {
    const float* x  = (const float*)d_in[0];
    const int*   ei = (const int*)d_in[1];
    const float* W1 = (const float*)d_in[2];
    const float* b1 = (const float*)d_in[3];
    const float* W2 = (const float*)d_in[4];
    const float* b2 = (const float*)d_in[5];
    const float* Wl = (const float*)d_in[6];
    const float* bl = (const float*)d_in[7];

    const int N = in_sizes[0] / 64;
    const int E = in_sizes[1] / 2;
    const int* src = ei;
    const int* dst = ei + E;

    // Workspace carve-out (all 256B aligned): ~58 MB total
    char*  ws  = (char*)d_ws;
    size_t off = 0;
    auto carve = [&](size_t bytes) -> void* {
        void* p = ws + off;
        off += (bytes + 255) & ~(size_t)255;
        return p;
    };
    float* inv  = (float*)carve((size_t)N * 4);        // deg -> deg^{-1/2}
    float* norm = (float*)carve((size_t)E * 4);        // per-edge norm
    float* bufA = (float*)carve((size_t)N * 64 * 4);   // hlin
    float* bufB = (float*)carve((size_t)N * 64 * 4);   // agg / h

    const int n16 = N * 16;
    const int gN    = (N + THREADS - 1) / THREADS;
    const int gE    = (E + THREADS - 1) / THREADS;
    const int gN16  = (n16 + THREADS - 1) / THREADS;
    const int gE16  = (int)(((size_t)E * 16 + THREADS - 1) / THREADS);
    const int tiles = (N + 15) / 16;
    const int gGemm = (tiles + 7) / 8;                 // 8 waves / 256-thread block

    // normalization prep
    k_init_deg<<<gN, THREADS, 0, stream>>>(inv, N);
    k_count_deg<<<gE, THREADS, 0, stream>>>(dst, inv, E);
    k_inv<<<gN, THREADS, 0, stream>>>(inv, N);
    k_edge_norm<<<gE, THREADS, 0, stream>>>(src, dst, inv, norm, E);

    // layer 1
    k_gemm_wmma<64, false><<<gGemm, THREADS, 0, stream>>>(x, W1, nullptr, bufA, N);
    k_self_init<<<gN16, THREADS, 0, stream>>>(bufA, inv, bufB, n16);
    k_scatter<<<gE16, THREADS, 0, stream>>>(src, dst, norm, bufA, bufB, E);
    k_bias_relu<<<gN16, THREADS, 0, stream>>>(bufB, b1, n16);

    // layer 2
    k_gemm_wmma<64, false><<<gGemm, THREADS, 0, stream>>>(bufB, W2, nullptr, bufA, N);
    k_self_init<<<gN16, THREADS, 0, stream>>>(bufA, inv, bufB, n16);
    k_scatter<<<gE16, THREADS, 0, stream>>>(src, dst, norm, bufA, bufB, E);
    k_bias_relu<<<gN16, THREADS, 0, stream>>>(bufB, b2, n16);

    // final projection (fused bias) -> d_out
    k_gemm_wmma<16, true><<<gGemm, THREADS, 0, stream>>>(bufB, Wl, bl, (float*)d_out, N);
}